// SimpleMaskModule_32564442038291
// MI455X (gfx1250) — compile-verified
//
#include <hip/hip_runtime.h>
#include <cstdint>
#include <cstddef>

#define DIM_REP 2048
#define DIM_VEC 64
#define BM      64            // tile rows/cols per block (symmetric launch -> keep blocks plentiful)
#define LDSW    68            // padded row stride in floats: bank = (4r+k)%64 -> conflict-free b128
#define TS      66            // transpose buffer stride: bank = (2r+c)%64 -> conflict-free b32

// Generic pointer to __shared__ object -> raw LDS byte offset
// (generic->local addrspacecast lowers to taking the low 32 bits).
__device__ __forceinline__ unsigned lds_offset(const void* p) {
    return (unsigned)(uintptr_t)(const __attribute__((address_space(3))) void*)p;
}

__global__ __launch_bounds__(256) void l1mask_kernel(const float* __restrict__ lam,
                                                     float* __restrict__ out) {
    __shared__ float As[BM * LDSW];
    __shared__ float Bs[BM * LDSW];

    const int bx = blockIdx.x;    // column tile (j)
    const int by = blockIdx.y;    // row tile (i)
    if (bx < by) return;          // symmetry: mask[i,j] == mask[j,i]; compute upper triangle only

    const int tid = threadIdx.x;

    // ---------------------------------------------------------------
    // Stage A (i-rows) and B (j-rows) tiles: 64 x 64 f32 = 16KB each,
    // via gfx1250 async global->LDS DMA (ASYNCcnt path).
    // Tile source is contiguous: 1024 x 16B chunks; 4 chunks/thread.
    // ---------------------------------------------------------------
    const int q  = tid & 15;      // 16B chunk within a 64-float row
    const int r0 = tid >> 4;      // starting row
    const uint64_t gA = (uint64_t)(const void*)(lam + (size_t)by * BM * DIM_VEC);
    const uint64_t gB = (uint64_t)(const void*)(lam + (size_t)bx * BM * DIM_VEC);
#pragma unroll
    for (int s = 0; s < 4; ++s) {
        const int row = r0 + 16 * s;
        const unsigned la = lds_offset(&As[row * LDSW]) + (unsigned)(q * 16);
        const unsigned lb = lds_offset(&Bs[row * LDSW]) + (unsigned)(q * 16);
        const uint64_t ga = gA + (uint64_t)(row * (DIM_VEC * 4) + q * 16);
        const uint64_t gb = gB + (uint64_t)(row * (DIM_VEC * 4) + q * 16);
        asm volatile("global_load_async_to_lds_b128 %0, %1, off"
                     :: "v"(la), "v"(ga) : "memory");
        asm volatile("global_load_async_to_lds_b128 %0, %1, off"
                     :: "v"(lb), "v"(gb) : "memory");
    }
    asm volatile("s_wait_asynccnt 0x0" ::: "memory");
    __syncthreads();

    // ---------------------------------------------------------------
    // 4x4 micro-tile per thread, interleaved by 16: lane-adjacent
    // threads touch adjacent columns (coalesced stores, bank-spread
    // B reads, broadcast A reads).
    // ---------------------------------------------------------------
    const int tx = tid & 15;      // column lane
    const int ty = tid >> 4;      // row lane

    float acc[4][4];
#pragma unroll
    for (int u = 0; u < 4; ++u)
#pragma unroll
        for (int v = 0; v < 4; ++v) acc[u][v] = 0.0f;

    for (int kk = 0; kk < DIM_VEC; kk += 4) {
        float4 a4[4], b4[4];
#pragma unroll
        for (int u = 0; u < 4; ++u)
            a4[u] = *(const float4*)&As[(ty + 16 * u) * LDSW + kk];
#pragma unroll
        for (int v = 0; v < 4; ++v)
            b4[v] = *(const float4*)&Bs[(tx + 16 * v) * LDSW + kk];
#pragma unroll
        for (int u = 0; u < 4; ++u)
#pragma unroll
            for (int v = 0; v < 4; ++v) {
                // sub + add-with-|src| modifier: 2 VALU / element, VOPD-friendly
                acc[u][v] += fabsf(a4[u].x - b4[v].x);
                acc[u][v] += fabsf(a4[u].y - b4[v].y);
                acc[u][v] += fabsf(a4[u].z - b4[v].z);
                acc[u][v] += fabsf(a4[u].w - b4[v].w);
            }
    }

    // mask = exp(-L1)
    float e[4][4];
#pragma unroll
    for (int u = 0; u < 4; ++u)
#pragma unroll
        for (int v = 0; v < 4; ++v) e[u][v] = __expf(-acc[u][v]);

    // ---------------------------------------------------------------
    // Direct tile store: rows by*64+ty+16u, cols bx*64+tx+16v
    // (lanes 0..15 hit adjacent columns -> coalesced).
    // ---------------------------------------------------------------
    const size_t obase = ((size_t)by * BM + ty) * DIM_REP + (size_t)bx * BM + tx;
#pragma unroll
    for (int u = 0; u < 4; ++u) {
        const size_t orow = obase + (size_t)(16 * u) * DIM_REP;
#pragma unroll
        for (int v = 0; v < 4; ++v)
            out[orow + 16 * v] = e[u][v];
    }

    // ---------------------------------------------------------------
    // Mirrored tile (bx,by swapped) via LDS transpose so the global
    // stores stay coalesced. Reuse As as 64x66 buffer (16.9KB <= 17.4KB).
    // ---------------------------------------------------------------
    if (bx != by) {
        __syncthreads();              // everyone done reading As/Bs
        float* T = As;
#pragma unroll
        for (int u = 0; u < 4; ++u)
#pragma unroll
            for (int v = 0; v < 4; ++v)
                T[(tx + 16 * v) * TS + (ty + 16 * u)] = e[u][v];   // T[j_local][i_local]
        __syncthreads();

        const size_t tbase = ((size_t)bx * BM + ty) * DIM_REP + (size_t)by * BM + tx;
#pragma unroll
        for (int u = 0; u < 4; ++u) {
            const size_t trow = tbase + (size_t)(16 * u) * DIM_REP;
#pragma unroll
            for (int v = 0; v < 4; ++v)
                out[trow + 16 * v] = T[(ty + 16 * u) * TS + (tx + 16 * v)];
        }
    }
}

// Tail of d_out = lambdas passthrough (second tuple element): 131072 f32.
__global__ __launch_bounds__(256) void copy_lambdas_kernel(const float* __restrict__ lam,
                                                           float* __restrict__ out) {
    const int i = blockIdx.x * 256 + threadIdx.x;   // 32768 float4 chunks
    ((float4*)out)[i] = ((const float4*)lam)[i];
}

extern "C" void kernel_launch(void* const* d_in, const int* in_sizes, int n_in,
                              void* d_out, int out_size, void* d_ws, size_t ws_size,
                              hipStream_t stream) {
    (void)in_sizes; (void)n_in; (void)out_size; (void)d_ws; (void)ws_size;
    const float* lam = (const float*)d_in[0];
    float* out = (float*)d_out;

    dim3 grid(DIM_REP / BM, DIM_REP / BM);      // 32 x 32; lower-triangle blocks exit immediately
    l1mask_kernel<<<grid, 256, 0, stream>>>(lam, out);

    // lambdas copied after the 2048x2048 mask
    copy_lambdas_kernel<<<(DIM_REP * DIM_VEC / 4) / 256, 256, 0, stream>>>(
        lam, out + (size_t)DIM_REP * DIM_REP);
}